// gaussian_3d_volume_42906723287366
// MI455X (gfx1250) — compile-verified
//
#include <hip/hip_runtime.h>
#include <hip/hip_bf16.h>
#include <math.h>

#define N_POINTS 65536
#define N_GAUSS  1024
#define N_CHUNKS (N_GAUSS / 16)   // 64 chunks of 16 gaussians
// per chunk layout in workspace: 3 row-sets * 64 floats (WMMA-B lane order) = 192 floats
#define CHUNK_FLOATS 192

typedef float v2f __attribute__((ext_vector_type(2)));
typedef float v8f __attribute__((ext_vector_type(8)));

#if __has_builtin(__builtin_amdgcn_exp2f)
#define EXP2F(x) __builtin_amdgcn_exp2f(x)
#else
#define EXP2F(x) exp2f(x)
#endif

// kappa = sqrt(0.5 * log2(e)); folding it into A and b turns exp(-0.5*q) into exp2(-||A p + b||^2)
#define KAPPA 0.84932213f
#define PI_F  3.14159265358979323846f

// ---------------------------------------------------------------------------
// Kernel 1: per-Gaussian preprocessing (1024 threads total, negligible cost).
// Computes A = diag(kappa/s) * (Rz*Ry*Rx), b = -A*c, and scatters rows into
// the exact per-lane layout the WMMA B operand wants:
//   chunk c, row-set r, lane L reads float2 at W[c*192 + r*64 + 2*L]
//   lanes 0..15  (gaussian j=L)    hold (A[r][0], A[r][1])   -> K0,K1
//   lanes 16..31 (gaussian j=L-16) hold (A[r][2], b[r])      -> K2,K3
// Alphas are prescaled by 1/N_GAUSS so the final mean is free.
// ---------------------------------------------------------------------------
__global__ __launch_bounds__(256)
void g3d_prep_kernel(const float* __restrict__ centers,
                     const float* __restrict__ angles,
                     const float* __restrict__ scales,
                     const float* __restrict__ alphas,
                     float* __restrict__ W,
                     float* __restrict__ alph) {
    int m = blockIdx.x * blockDim.x + threadIdx.x;
    if (m >= N_GAUSS) return;

    // deg2rad(angle * 180) == angle * pi
    float ax = angles[3 * m + 0] * PI_F;
    float ay = angles[3 * m + 1] * PI_F;
    float az = angles[3 * m + 2] * PI_F;
    float sx = __sinf(ax), cx = __cosf(ax);
    float sy = __sinf(ay), cy = __cosf(ay);
    float sz = __sinf(az), cz = __cosf(az);

    // R = Rz * Ry * Rx  (ZYX Euler)
    float R[3][3];
    R[0][0] = cz * cy;  R[0][1] = -sz * cx + cz * sy * sx;  R[0][2] =  sz * sx + cz * sy * cx;
    R[1][0] = sz * cy;  R[1][1] =  cz * cx + sz * sy * sx;  R[1][2] = -cz * sx + sz * sy * cx;
    R[2][0] = -sy;      R[2][1] =  cy * sx;                 R[2][2] =  cy * cx;

    float c0 = centers[3 * m + 0], c1 = centers[3 * m + 1], c2 = centers[3 * m + 2];

    float A[3][3], b[3];
    #pragma unroll
    for (int i = 0; i < 3; ++i) {
        float inv = KAPPA / scales[3 * m + i];
        A[i][0] = R[i][0] * inv;
        A[i][1] = R[i][1] * inv;
        A[i][2] = R[i][2] * inv;
        b[i] = -(A[i][0] * c0 + A[i][1] * c1 + A[i][2] * c2);
    }

    int c = m >> 4;
    int j = m & 15;
    float* base = W + c * CHUNK_FLOATS;
    #pragma unroll
    for (int r = 0; r < 3; ++r) {
        base[r * 64 + 2 * j + 0]      = A[r][0];   // lane j,    K0
        base[r * 64 + 2 * j + 1]      = A[r][1];   // lane j,    K1
        base[r * 64 + 32 + 2 * j + 0] = A[r][2];   // lane j+16, K2
        base[r * 64 + 32 + 2 * j + 1] = b[r];      // lane j+16, K3
    }
    alph[m] = alphas[m] * (1.0f / (float)N_GAUSS);
}

// ---------------------------------------------------------------------------
// Kernel 2: main evaluation.
// One wave32 owns a tile of 16 points; A operand (points, homogeneous coord)
// is built once and reused by all 64 chunks * 3 WMMAs. Per chunk:
//   d0/d1/d2 = WMMA(A, B_r) give scaled local coords, elementwise aligned in
//   the C/D layout, so q = d0^2+d1^2+d2^2 is pure per-lane VALU, then
//   acc += alpha * exp2(-q). Final 16-lane xor-shuffle reduction; lanes 0/16
//   store two float4 each (points are contiguous within the tile).
// ---------------------------------------------------------------------------
__global__ __launch_bounds__(256)
void g3d_eval_kernel(const float* __restrict__ pts,
                     const float* __restrict__ W,
                     const float* __restrict__ alph,
                     float* __restrict__ out) {
    const int lane = threadIdx.x & 31;
    const int wave = threadIdx.x >> 5;
    const int tile = blockIdx.x * 8 + wave;          // 16 points per wave
    const int prow = tile * 16 + (lane & 15);
    const bool hi = lane >= 16;

    // A operand: 16x4 f32. lanes 0-15: (K0,K1)=(x,y); lanes 16-31: (K2,K3)=(z,1)
    const float* p = pts + 3 * prow;
    float px = p[0], py = p[1], pz = p[2];
    v2f a;
    a.x = hi ? pz : px;
    a.y = hi ? 1.0f : py;

    const float* wl = W + 2 * lane;                  // per-lane B base
    const float* al = alph + (lane & 15);

    v8f acc = {0.f, 0.f, 0.f, 0.f, 0.f, 0.f, 0.f, 0.f};
    const v8f zc  = {0.f, 0.f, 0.f, 0.f, 0.f, 0.f, 0.f, 0.f};

    for (int c = 0; c < N_CHUNKS; ++c) {
        const float* blk = wl + c * CHUNK_FLOATS;
        v2f b0 = *(const v2f*)(blk);
        v2f b1 = *(const v2f*)(blk + 64);
        v2f b2 = *(const v2f*)(blk + 128);
        float alpha = al[c * 16];

        v8f d0 = __builtin_amdgcn_wmma_f32_16x16x4_f32(false, a, false, b0,
                                                       (short)0, zc, false, false);
        v8f d1 = __builtin_amdgcn_wmma_f32_16x16x4_f32(false, a, false, b1,
                                                       (short)0, zc, false, false);
        v8f d2 = __builtin_amdgcn_wmma_f32_16x16x4_f32(false, a, false, b2,
                                                       (short)0, zc, false, false);

        #pragma unroll
        for (int r = 0; r < 8; ++r) {
            // q = -(d0^2 + d1^2 + d2^2): neg folds into FMA source modifiers
            float q = d0[r] * -d0[r];
            q = fmaf(d1[r], -d1[r], q);
            q = fmaf(d2[r], -d2[r], q);
            acc[r] = fmaf(alpha, EXP2F(q), acc[r]);
        }
    }

    // Reduce each accumulator across the 16 columns (two independent
    // 16-lane groups under xor masks 1,2,4,8 in wave32).
    float red[8];
    #pragma unroll
    for (int r = 0; r < 8; ++r) {
        float v = acc[r];
        v += __shfl_xor(v, 1, 32);
        v += __shfl_xor(v, 2, 32);
        v += __shfl_xor(v, 4, 32);
        v += __shfl_xor(v, 8, 32);
        red[r] = v;
    }

    if ((lane & 15) == 0) {
        // lane 0 -> points tile*16+0..7, lane 16 -> points tile*16+8..15
        float* o = out + tile * 16 + (hi ? 8 : 0);
        float4 lo4 = make_float4(red[0], red[1], red[2], red[3]);
        float4 hi4 = make_float4(red[4], red[5], red[6], red[7]);
        *(float4*)(o + 0) = lo4;
        *(float4*)(o + 4) = hi4;
    }
}

// ---------------------------------------------------------------------------
extern "C" void kernel_launch(void* const* d_in, const int* in_sizes, int n_in,
                              void* d_out, int out_size, void* d_ws, size_t ws_size,
                              hipStream_t stream) {
    const float* points  = (const float*)d_in[0];
    const float* centers = (const float*)d_in[1];
    const float* angles  = (const float*)d_in[2];
    const float* scales  = (const float*)d_in[3];
    const float* alphas  = (const float*)d_in[4];

    float* W    = (float*)d_ws;                   // 64 * 192 = 12288 floats
    float* alph = W + N_CHUNKS * CHUNK_FLOATS;    // + 1024 floats  (~53 KB total)

    g3d_prep_kernel<<<(N_GAUSS + 255) / 256, 256, 0, stream>>>(
        centers, angles, scales, alphas, W, alph);

    // 16 points/wave * 8 waves/block = 128 points/block -> 512 blocks
    g3d_eval_kernel<<<N_POINTS / 128, 256, 0, stream>>>(
        points, W, alph, (float*)d_out);
}